// DetectionLayer_24008867184919
// MI455X (gfx1250) — compile-verified
//
#include <hip/hip_runtime.h>
#include <hip/hip_bf16.h>

#define BB   16
#define NN   2000
#define CC   81
#define META_STRIDE (12 + CC)
#define MAX_INST 100
#define MIN_CONF 0.7f
#define NMS_THR  0.3f
#define NEGV    (-1000000000.0f)
#define NPAD 2048
// per-batch staging block in d_ws (floats): [scores 2000][boxes 8000][cls 2000]
#define WSB 12000

typedef unsigned int u32x4 __attribute__((ext_vector_type(4)));
typedef int          i32x4 __attribute__((ext_vector_type(4)));
typedef int          i32x8 __attribute__((ext_vector_type(8)));

// ---------------------------------------------------------------------------
// Tensor Data Mover: 1-D global -> LDS copy of `ndwords` DWORDs.
// D# built per CDNA5 ISA ch.8: group0 {count=1, lds_addr, global_addr, type=2},
// group1 {data_size=4B, tensor_dim0=ndwords, tensor_dim1=1, tile_dim0=ndwords,
//         tile_dim1=1, stride0=ndwords}. Groups 2/3 unused (2-D tensor).
// ---------------------------------------------------------------------------
__device__ __forceinline__ void tdm_load_1d(const void* gptr, unsigned lds_off,
                                            unsigned ndwords) {
  unsigned long long ga = (unsigned long long)gptr;
  u32x4 g0;
  g0[0] = 1u;                                    // count=1, is_restore=0
  g0[1] = lds_off;                               // lds_addr (bytes)
  g0[2] = (unsigned)(ga & 0xFFFFFFFFu);          // global_addr[31:0]
  g0[3] = (unsigned)((ga >> 32) & 0x1FFFFFFu)    // global_addr[56:32]
          | (2u << 30);                          // type=2 ("image")
  i32x8 g1;
  g1[0] = (2 << 16);                             // data_size=2 -> 4 bytes
  g1[1] = (int)((ndwords & 0xFFFFu) << 16);      // tensor_dim0[15:0]
  g1[2] = (int)((ndwords >> 16) & 0xFFFFu)       // tensor_dim0[31:16]
          | (1 << 16);                           // tensor_dim1 = 1
  g1[3] = (int)((ndwords & 0xFFFFu) << 16);      // tile_dim0 = ndwords (<=8000)
  g1[4] = 1;                                     // tile_dim1=1, tile_dim2=0
  g1[5] = (int)ndwords;                          // tensor_dim0_stride lo32
  g1[6] = 0;                                     // stride0 hi / stride1 lo
  g1[7] = 0;
  i32x4 z4 = {0, 0, 0, 0};
#if __clang_major__ >= 23
  i32x8 z8 = {0, 0, 0, 0, 0, 0, 0, 0};
  __builtin_amdgcn_tensor_load_to_lds(g0, g1, z4, z4, z8, 0);
#else
  __builtin_amdgcn_tensor_load_to_lds(g0, g1, z4, z4, 0);
#endif
}

// Flat shared pointer -> LDS byte offset (LDS offset is the low 32 bits of the address).
__device__ __forceinline__ unsigned lds_off(const void* p) {
  return (unsigned)(unsigned long long)p;
}

// ---------------------------------------------------------------------------
// Kernel 1: per-proposal argmax + delta apply + clip + confidence mask.
// Writes per-batch staging block: [masked_score 2000][boxes 2000x4][cls 2000].
// ---------------------------------------------------------------------------
__global__ __launch_bounds__(256) void det_refine_kernel(
    const float* __restrict__ rois, const float* __restrict__ probs,
    const float* __restrict__ deltas, const float* __restrict__ meta,
    float* __restrict__ ws) {
  int t = blockIdx.x * blockDim.x + threadIdx.x;
  if (t >= BB * NN) return;
  int b = t / NN, n = t % NN;

  const float* p = probs + (size_t)(b * NN + n) * CC;
  __builtin_prefetch(p, 0, 1);                       // global_prefetch_b8
  const float* r = rois + (size_t)(b * NN + n) * 4;
  __builtin_prefetch(r, 0, 1);

  // argmax over classes (first max, matching jnp.argmax)
  int best = 0;
  float bs = p[0];
  #pragma unroll 3
  for (int c = 1; c < CC; ++c) {
    float v = p[c];
    if (v > bs) { bs = v; best = c; }
  }

  const float* d = deltas + ((size_t)(b * NN + n) * CC + best) * 4;
  float dy = d[0] * 0.1f, dx = d[1] * 0.1f;
  float dh = d[2] * 0.2f, dw = d[3] * 0.2f;

  float y1 = r[0], x1 = r[1], y2 = r[2], x2 = r[3];
  float h = y2 - y1, w = x2 - x1;
  float cy = y1 + 0.5f * h + dy * h;
  float cx = x1 + 0.5f * w + dx * w;
  h *= expf(dh);
  w *= expf(dw);
  float ny1 = cy - 0.5f * h, nx1 = cx - 0.5f * w;
  float ny2 = cy + 0.5f * h, nx2 = cx + 0.5f * w;

  // normalized window: shape from meta row 0, window from meta row b
  float H = meta[4], W = meta[5];
  const float* mb = meta + (size_t)b * META_STRIDE;
  float sy = 1.0f / (H - 1.0f), sx = 1.0f / (W - 1.0f);
  float wy1 = mb[7] * sy;
  float wx1 = mb[8] * sx;
  float wy2 = (mb[9] - 1.0f) * sy;
  float wx2 = (mb[10] - 1.0f) * sx;

  ny1 = fminf(fmaxf(ny1, wy1), wy2);
  nx1 = fminf(fmaxf(nx1, wx1), wx2);
  ny2 = fminf(fmaxf(ny2, wy1), wy2);
  nx2 = fminf(fmaxf(nx2, wx1), wx2);

  bool kept = (best > 0) && (bs >= MIN_CONF);
  float ms = kept ? bs : NEGV;

  float* base = ws + (size_t)b * WSB;
  base[n] = ms;
  base[NN + 4 * n + 0] = ny1;
  base[NN + 4 * n + 1] = nx1;
  base[NN + 4 * n + 2] = ny2;
  base[NN + 4 * n + 3] = nx2;
  ((int*)(base + 5 * NN))[n] = best;
}

// ---------------------------------------------------------------------------
// Kernel 2: one workgroup per batch. TDM-stage the 48KB batch block into LDS,
// bitonic-sort (score desc), greedy class-aware NMS, ordered compaction.
// ---------------------------------------------------------------------------
__global__ __launch_bounds__(1024) void det_nms_kernel(
    const float* __restrict__ ws, float* __restrict__ out) {
  __shared__ float s_key[NPAD];        //  8192 B  sorted masked scores
  __shared__ int   s_idx[NPAD];        //  8192 B  original proposal index
  __shared__ float s_box[NN * 4];      // 32000 B  boxes (original order)
  __shared__ int   s_cls[NN];          //  8000 B  class ids (original order)
  __shared__ unsigned char s_sup[NPAD];//  2048 B  suppression flags
  __shared__ int s_count;

  const int b = blockIdx.x;
  const int tid = threadIdx.x;
  const float* src = ws + (size_t)b * WSB;

  // --- async TDM staging (wave 0 issues; TDM ignores EXEC) ---
  if (tid == 0) {
    tdm_load_1d(src,          lds_off(s_key), NN);       // scores
    tdm_load_1d(src + NN,     lds_off(s_box), NN * 4);   // boxes
    tdm_load_1d(src + 5 * NN, lds_off(s_cls), NN);       // classes
  }
  // pad + identity permutation while the DMA is in flight
  for (int t = tid; t < NPAD; t += 1024) {
    s_idx[t] = t;
    if (t >= NN) s_key[t] = NEGV;
  }
  if (tid == 0) __builtin_amdgcn_s_wait_tensorcnt(0);
  __syncthreads();

  // --- bitonic sort, descending by key, payload = idx ---
  for (int k = 2; k <= NPAD; k <<= 1) {
    for (int j = k >> 1; j > 0; j >>= 1) {
      for (int t = tid; t < NPAD; t += 1024) {
        int q = t ^ j;
        if (q > t) {
          bool desc = ((t & k) == 0);
          float ka = s_key[t], kb = s_key[q];
          if (desc ? (ka < kb) : (ka > kb)) {
            s_key[t] = kb; s_key[q] = ka;
            int ia = s_idx[t]; s_idx[t] = s_idx[q]; s_idx[q] = ia;
          }
        }
      }
      __syncthreads();
    }
  }

  // --- initial suppression: everything below confidence ---
  for (int t = tid; t < NPAD; t += 1024)
    s_sup[t] = (s_key[t] < MIN_CONF) ? 1 : 0;
  __syncthreads();

  // --- greedy class-aware NMS over sorted order ---
  for (int i = 0; i < NN; ++i) {
    if (!s_sup[i]) {
      int oi = s_idx[i];
      int ci = s_cls[oi];
      float by1 = s_box[4 * oi + 0], bx1 = s_box[4 * oi + 1];
      float by2 = s_box[4 * oi + 2], bx2 = s_box[4 * oi + 3];
      float a1 = (by2 - by1) * (bx2 - bx1);
      for (int t = tid; t < NN; t += 1024) {
        if (t > i && !s_sup[t]) {
          int ot = s_idx[t];
          if (s_cls[ot] == ci) {
            float cy1 = s_box[4 * ot + 0], cx1 = s_box[4 * ot + 1];
            float cy2 = s_box[4 * ot + 2], cx2 = s_box[4 * ot + 3];
            float iy1 = fmaxf(by1, cy1), ix1 = fmaxf(bx1, cx1);
            float iy2 = fminf(by2, cy2), ix2 = fminf(bx2, cx2);
            float inter = fmaxf(iy2 - iy1, 0.0f) * fmaxf(ix2 - ix1, 0.0f);
            float a2 = (cy2 - cy1) * (cx2 - cx1);
            float iou = inter / (a1 + a2 - inter + 1e-12f);
            if (iou > NMS_THR) s_sup[t] = 1;
          }
        }
      }
    }
    __syncthreads();
  }

  if (tid == 0) s_count = 0;
  __syncthreads();

  // --- ordered compaction of survivors into the top-100 slots ---
  for (int t = tid; t < NN; t += 1024) {
    if (!s_sup[t]) {
      int rank = 0;
      for (int k2 = 0; k2 < t; ++k2) rank += s_sup[k2] ? 0 : 1;
      atomicAdd(&s_count, 1);
      if (rank < MAX_INST) {
        int o = s_idx[t];
        float* dst = out + ((size_t)b * MAX_INST + rank) * 6;
        dst[0] = s_box[4 * o + 0];
        dst[1] = s_box[4 * o + 1];
        dst[2] = s_box[4 * o + 2];
        dst[3] = s_box[4 * o + 3];
        dst[4] = (float)s_cls[o];
        dst[5] = s_key[t];
      }
    }
  }
  __syncthreads();

  // zero the unfilled tail (each slot written by exactly one thread)
  int cnt = s_count; if (cnt > MAX_INST) cnt = MAX_INST;
  for (int r = tid; r < MAX_INST; r += 1024) {
    if (r >= cnt) {
      float* dst = out + ((size_t)b * MAX_INST + r) * 6;
      dst[0] = 0.0f; dst[1] = 0.0f; dst[2] = 0.0f;
      dst[3] = 0.0f; dst[4] = 0.0f; dst[5] = 0.0f;
    }
  }
}

extern "C" void kernel_launch(void* const* d_in, const int* in_sizes, int n_in,
                              void* d_out, int out_size, void* d_ws, size_t ws_size,
                              hipStream_t stream) {
  const float* rois  = (const float*)d_in[0];   // (B,N,4)
  const float* probs = (const float*)d_in[1];   // (B,N,C)
  const float* bbox  = (const float*)d_in[2];   // (B,N,C,4)
  const float* meta  = (const float*)d_in[3];   // (B,12+C)
  float* out = (float*)d_out;                   // (B,100,6)
  float* ws  = (float*)d_ws;                    // >= B*12000 floats

  int total = BB * NN;
  det_refine_kernel<<<(total + 255) / 256, 256, 0, stream>>>(rois, probs, bbox,
                                                             meta, ws);
  det_nms_kernel<<<BB, 1024, 0, stream>>>(ws, out);
}